// relation_SSD_68186900791974
// MI455X (gfx1250) — compile-verified
//
#include <hip/hip_runtime.h>

typedef __attribute__((ext_vector_type(2))) float v2f;
typedef __attribute__((ext_vector_type(8))) float v8f;

#define B_   16
#define L_   1444
#define D_   128
#define C_   21
#define KT   32     // K-block staged in LDS
#define MT   32     // M rows per workgroup (two 16-row WMMA subtiles)
#define BSTR 136    // LDS row stride (floats): 2*136 % 64 == 16 -> K-half lane groups on disjoint banks

// ---------------------------------------------------------------------------
// One K-block of WMMA accumulation.
// MODE 0: every j in block > every i in tile  -> g = P[cj, ci]   (uniform)
// MODE 1: every j in block < every i in tile  -> g = P[ci, cj]   (uniform)
// MODE 2: general (diagonal overlap and/or K tail padding), fully branchless
// ---------------------------------------------------------------------------
template <int MODE>
__device__ __forceinline__ void k_block(
    const float* Ps, const float (*Bs)[BSTR], const int* cjs,
    int k0, int i, int ci, int ciRow, float dval, int khalf, int nA, int nB,
    v8f& acc0, v8f& acc1)
{
    #pragma unroll
    for (int kk = 0; kk < KT; kk += 4) {
        const int jl0 = kk + khalf * 2;   // this lane's first local K index
        // ---- A fragment: graph entries generated in-register ----
        v2f a;
        #pragma unroll
        for (int u = 0; u < 2; ++u) {
            const int jl = jl0 + u;
            const int cj = cjs[jl];
            float g;
            if (MODE == 0) {
                g = Ps[cj * C_ + ci];                   // upper triangle, mirrored
            } else if (MODE == 1) {
                g = Ps[ciRow + cj];                     // lower triangle
            } else {
                const int j     = k0 + jl;
                const bool low  = (i > j);
                const int  rsel = low ? ci : cj;        // v_cndmask
                const int  csel = low ? cj : ci;        // v_cndmask
                g = Ps[rsel * C_ + csel];               // v_mad + ds_load (unconditional)
                g = (i == j) ? dval : g;                // diagonal fixup
                g = (j >= L_) ? 0.0f : g;               // K padding fixup
            }
            a[u] = g;
        }
        // ---- B fragments from LDS ----
        v2f b0, b1;
        b0.x = Bs[jl0][nA];  b0.y = Bs[jl0 + 1][nA];
        b1.x = Bs[jl0][nB];  b1.y = Bs[jl0 + 1][nB];
        acc0 = __builtin_amdgcn_wmma_f32_16x16x4_f32(
                   false, a, false, b0, (short)0, acc0, false, false);
        acc1 = __builtin_amdgcn_wmma_f32_16x16x4_f32(
                   false, a, false, b1, (short)0, acc1, false, false);
    }
}

// ---------------------------------------------------------------------------
// fused = graph @ source; graph[i,j] = f(class_idx[i], class_idx[j], i<=>j)
// grid.x = B * ceil(L/32); block = 256 threads (8 waves).
// wave w: M-subtile (w>>2)*16, N-slice (w&3)*32 (two 16-wide WMMA tiles).
// ---------------------------------------------------------------------------
__global__ __launch_bounds__(256) void relation_graph_gemm(
    const float* __restrict__ P,      // [21*21] cls_r_prob
    const float* __restrict__ src,    // [B*L*D]
    const int*   __restrict__ idx,    // [B*L] class_idx
    float*       __restrict__ fused)  // [B*L*D]
{
    __shared__ float Ps[C_ * C_];
    __shared__ __align__(16) float Bs[KT][BSTR];
    __shared__ int   cjs[KT];

    const int tid  = threadIdx.x;
    const int lane = tid & 31;
    const int wave = tid >> 5;

    const int tilesPerBatch = (L_ + MT - 1) / MT;      // 46
    const int b  = blockIdx.x / tilesPerBatch;
    const int i0 = (blockIdx.x % tilesPerBatch) * MT;

    for (int t = tid; t < C_ * C_; t += 256) Ps[t] = P[t];

    const int   mi    = lane & 15;
    const int   khalf = lane >> 4;
    const int   sub   = wave >> 2;                     // 16-row subtile within M-tile
    const int   n0    = (wave & 3) * 32;
    const int   i     = i0 + sub * 16 + mi;            // output row this lane builds A for
    const int   iC    = (i < L_) ? i : (L_ - 1);
    const int   ci    = idx[(size_t)b * L_ + iC];
    const int   ciRow = ci * C_;
    const float dval  = (ci != 0) ? 1.0f : 0.0f;       // diagonal value for row i
    const int   nA    = n0 + mi;
    const int   nB    = n0 + 16 + mi;

    const float* srcB = src + (size_t)b * L_ * D_;

    v8f acc0 = {};
    v8f acc1 = {};

    for (int k0 = 0; k0 < L_; k0 += KT) {
        const bool full = (k0 + KT <= L_);             // only last block is partial
        __syncthreads();
        if (tid < KT) {
            const int j = k0 + tid;
            cjs[tid] = (full || j < L_) ? idx[(size_t)b * L_ + j] : 0;
        }
        if (full) {
            // unguarded staging: 32x128 floats, 256 threads -> 4 float4 each
            #pragma unroll
            for (int t = tid; t < KT * (D_ / 4); t += 256) {
                const int r  = t >> 5;                 // D_/4 == 32
                const int c4 = (t & 31) * 4;
                *(float4*)&Bs[r][c4] =
                    *(const float4*)(srcB + (size_t)(k0 + r) * D_ + c4);
            }
        } else {
            #pragma unroll
            for (int t = tid; t < KT * (D_ / 4); t += 256) {
                const int r  = t >> 5;
                const int c4 = (t & 31) * 4;
                float4 v = make_float4(0.f, 0.f, 0.f, 0.f);
                if (k0 + r < L_)
                    v = *(const float4*)(srcB + (size_t)(k0 + r) * D_ + c4);
                *(float4*)&Bs[r][c4] = v;
            }
        }
        // prefetch next K block into cache while computing on this one
        {
            const int jn = k0 + KT + (tid & (KT - 1));
            if (jn < L_)
                __builtin_prefetch(srcB + (size_t)jn * D_ + (tid >> 5) * 16, 0, 1);
        }
        __syncthreads();

        if (full && k0 >= i0 + MT) {
            k_block<0>(Ps, Bs, cjs, k0, i, ci, ciRow, dval, khalf, nA, nB, acc0, acc1);
        } else if (k0 + KT <= i0) {
            k_block<1>(Ps, Bs, cjs, k0, i, ci, ciRow, dval, khalf, nA, nB, acc0, acc1);
        } else {
            k_block<2>(Ps, Bs, cjs, k0, i, ci, ciRow, dval, khalf, nA, nB, acc0, acc1);
        }
    }

    // C/D layout: VGPR r -> M = r + 8*khalf, N = lane&15
    #pragma unroll
    for (int r = 0; r < 8; ++r) {
        const int row = i0 + sub * 16 + r + 8 * khalf;
        if (row < L_) {
            const size_t base = ((size_t)b * L_ + row) * D_;
            fused[base + nA] = acc0[r];
            fused[base + nB] = acc1[r];
        }
    }
}

// ---------------------------------------------------------------------------
// second output: straight copy of source
// ---------------------------------------------------------------------------
__global__ void relation_copy_src(const float4* __restrict__ in,
                                  float4* __restrict__ out, int n4)
{
    const int t = blockIdx.x * blockDim.x + threadIdx.x;
    if (t < n4) out[t] = in[t];
}

extern "C" void kernel_launch(void* const* d_in, const int* in_sizes, int n_in,
                              void* d_out, int out_size, void* d_ws, size_t ws_size,
                              hipStream_t stream)
{
    const float* P   = (const float*)d_in[0];   // cls_r_prob [21,21]
    const float* src = (const float*)d_in[1];   // source [16,1444,128]
    const int*   idx = (const int*)d_in[2];     // class_idx [16,1444]

    float* fused   = (float*)d_out;                     // output 0
    float* src_out = fused + (size_t)B_ * L_ * D_;      // output 1

    const int tilesPerBatch = (L_ + MT - 1) / MT;       // 46
    dim3 grid(B_ * tilesPerBatch);
    relation_graph_gemm<<<grid, 256, 0, stream>>>(P, src, idx, fused);

    const int n4 = (B_ * L_ * D_) / 4;
    relation_copy_src<<<(n4 + 255) / 256, 256, 0, stream>>>(
        (const float4*)src, (float4*)src_out, n4);
}